// AGRAN_12421045420745
// MI455X (gfx1250) — compile-verified
//
#include <hip/hip_runtime.h>
#include <math.h>
#include <stdint.h>

typedef __attribute__((ext_vector_type(2))) float v2f;
typedef __attribute__((ext_vector_type(8))) float v8f;
typedef __attribute__((ext_vector_type(4))) unsigned v4u;
typedef __attribute__((ext_vector_type(8))) int v8i;
typedef __attribute__((ext_vector_type(4))) int v4i;

#define PADV (-4.294967295e9f)

#if defined(__AMDGCN__) && defined(__has_builtin)
#if __has_builtin(__builtin_amdgcn_tensor_load_to_lds) && \
    __has_builtin(__builtin_amdgcn_s_wait_tensorcnt)
#define AGR_TDM 1
#endif
#endif
#ifndef AGR_TDM
#define AGR_TDM 0
#endif

__device__ __forceinline__ v8f wmma4(v2f a, v2f b, v8f c) {
  // V_WMMA_F32_16X16X4_F32 : D = A(16x4) * B(4x16) + C(16x16)
  return __builtin_amdgcn_wmma_f32_16x16x4_f32(false, a, false, b, (short)0, c,
                                               false, false);
}

#if AGR_TDM
// TDM: DMA a 257x32-float tile (row stride 64 floats) from global into LDS.
// D# per CDNA5 ISA ch.8: group0 = {count=1, lds_addr, global_addr, type=2},
// group1 = {data_size=4B, tensor_dim0=32, tensor_dim1=257, tile_dim0=32,
//           tile_dim1=257, tensor_dim0_stride=64}; groups 2/3 zero (<=2D).
// This toolchain's builtin takes 6 args: (v4u, v8i, v4i, v4i, v8i, i32 cpol).
__device__ __forceinline__ void agr_tdm_table(const float* gsrc,
                                              unsigned lds_byte_off) {
  unsigned long long ga = (unsigned long long)(uintptr_t)gsrc;
  v4u g0 = {1u,                                     // count=1, user mode
            lds_byte_off,                           // lds_addr (bytes)
            (unsigned)(ga & 0xffffffffu),           // global_addr[31:0]
            (unsigned)((ga >> 32) & 0x01ffffffu) | 0x80000000u}; // [56:32] | type=2
  v8i g1 = {(int)0x00020000u,   // data_size=2 (4B), wg_mask=0
            (int)0x00200000u,   // tensor_dim0[15:0]=32 in bits[63:48]
            (int)0x01010000u,   // tensor_dim1[15:0]=257 in bits[95:80]
            (int)0x00200000u,   // tile_dim0=32 in bits[127:112]
            (int)0x00000101u,   // tile_dim1=257 in bits[143:128]; tile_dim2=0
            64,                 // tensor_dim0_stride[31:0]=64
            0, 0};
  v4i z4 = {0, 0, 0, 0};
  v8i z8 = {0, 0, 0, 0, 0, 0, 0, 0};
  __builtin_amdgcn_tensor_load_to_lds(g0, g1, z4, z4, z8, 0);
}
#endif

// ---------------------------------------------------------------------------
// seqs[b,l,:] = item_emb[log_seqs[b,l]] * sqrt(H) * (log!=0)
__global__ __launch_bounds__(256) void agr_embed(const int* __restrict__ logseq,
                                                 const float* __restrict__ item_emb,
                                                 float* __restrict__ seqs) {
  int e = blockIdx.x * 256 + threadIdx.x;       // < 409600
  int r = e >> 6, c = e & 63;
  int idx = logseq[r];
  seqs[e] = (idx == 0) ? 0.f : item_emb[idx * 64 + c] * 8.0f;
}

// ---------------------------------------------------------------------------
// Qn = LN(seqs); Q = Qn@Wq+bq; K = seqs@Wk+bk; V = seqs@Wv+bv
// 64 rows per workgroup, 256 threads (8 waves), WMMA f32 16x16x4.
__global__ __launch_bounds__(256) void agr_qkv(
    const float* __restrict__ seqs, float* __restrict__ Qn,
    float* __restrict__ Qb, float* __restrict__ Kb, float* __restrict__ Vb,
    const float* __restrict__ Wq, const float* __restrict__ bq,
    const float* __restrict__ Wk, const float* __restrict__ bk,
    const float* __restrict__ Wv, const float* __restrict__ bv,
    const float* __restrict__ lns, const float* __restrict__ lnb) {
  extern __shared__ float sm[];
  float* sX = sm;            // 64x64 raw seqs
  float* sQ = sm + 4096;     // 64x64 normalized
  float* sW = sm + 8192;     // 3 * 64x64 weights
  const int tid = threadIdx.x;
  const int r0 = blockIdx.x * 64;

  for (int e = tid; e < 4096; e += 256) {
    sX[e] = seqs[r0 * 64 + e];
    sW[e] = Wq[e];
    sW[4096 + e] = Wk[e];
    sW[8192 + e] = Wv[e];
  }
  __syncthreads();

  if (tid < 64) {
    float mu = 0.f, ss = 0.f;
    for (int c = 0; c < 64; ++c) { float v = sX[tid * 64 + c]; mu += v; ss += v * v; }
    mu *= (1.f / 64.f);
    float var = ss * (1.f / 64.f) - mu * mu;
    float rstd = rsqrtf(var + 1e-8f);
    for (int c = 0; c < 64; ++c) {
      float y = (sX[tid * 64 + c] - mu) * rstd * lns[c] + lnb[c];
      sQ[tid * 64 + c] = y;
      Qn[(r0 + tid) * 64 + c] = y;
    }
  }
  __syncthreads();

  const unsigned wv = tid >> 5, lane = tid & 31;
  const unsigned half = lane >> 4, lo = lane & 15, ko = half * 2;

  for (int j = wv; j < 48; j += 8) {               // 3 matrices * 16 tiles
    int m = j >> 4, t = j & 15, tr = t >> 2, tc = t & 3;
    const float* A = (m == 0) ? sQ : sX;
    const float* Bm = sW + m * 4096;
    const float* bias = (m == 0) ? bq : (m == 1 ? bk : bv);
    float* dst = (m == 0) ? Qb : (m == 1 ? Kb : Vb);
    v8f acc = {0.f, 0.f, 0.f, 0.f, 0.f, 0.f, 0.f, 0.f};
    for (int kk = 0; kk < 64; kk += 4) {
      v2f a, b;
      a.x = A[(tr * 16 + lo) * 64 + kk + ko];
      a.y = A[(tr * 16 + lo) * 64 + kk + ko + 1];
      b.x = Bm[(kk + ko) * 64 + tc * 16 + lo];
      b.y = Bm[(kk + ko + 1) * 64 + tc * 16 + lo];
      acc = wmma4(a, b, acc);
    }
    for (int v = 0; v < 8; ++v) {
      int row = r0 + tr * 16 + v + half * 8;
      int col = tc * 16 + lo;
      dst[row * 64 + col] = acc[v] + bias[col];
    }
  }
}

// ---------------------------------------------------------------------------
// Fused attention for one (b, head, 16-query tile).
// posK folded into K', posV into V'.  Rel-time/dis score terms via
// qt[q,t] = Q[q,:]·emb[t,:] tables; rel-V terms via per-(q,k) LDS gathers.
// The four 257x32 embedding tables are DMA'd into LDS by the Tensor Data
// Mover (wave 0, TENSORcnt), overlapping with the K'/V' VALU loads.
__global__ __launch_bounds__(256) void agr_attn(
    const float* __restrict__ Qg, const float* __restrict__ Kg,
    const float* __restrict__ Vg, const float* __restrict__ posK,
    const float* __restrict__ posV, const float* __restrict__ timeK,
    const float* __restrict__ timeV, const float* __restrict__ disK,
    const float* __restrict__ disV, const int* __restrict__ tmat,
    const int* __restrict__ dmat, const int* __restrict__ logseq,
    float* __restrict__ Out) {
  extern __shared__ float sm[];
  float* sQ  = sm;            // 16x32
  float* sK  = sm + 512;      // 208x32  (K + posK, zero padded)
  float* sV  = sm + 7168;     // 208x32  (V + posV)
  float* sS  = sm + 13824;    // 16x208  scores -> softmax A
  float* sQT = sm + 17152;    // 16x257  Q · timeK_emb^T
  float* sQD = sm + 21264;    // 16x257  Q · disK_emb^T
  float* sTK = sm + 25376;    // 257x32
  float* sDK = sm + 33600;    // 257x32
  float* sTV = sm + 41824;    // 257x32
  float* sDV = sm + 50048;    // 257x32
  float* sO  = sm + 58272;    // 16x32

  const int tid = threadIdx.x;
  int bid = blockIdx.x;
  const int qt = bid % 13;  bid /= 13;
  const int h  = bid & 1;   bid >>= 1;
  const int b  = bid;
  const int q0 = qt * 16;
  const int hc = h * 32;

#if AGR_TDM
  if (tid < 32) {            // wave 0: issue 4 tensor-DMA descriptors
    agr_tdm_table(timeK + hc, 4u * 25376u);
    agr_tdm_table(disK  + hc, 4u * 33600u);
    agr_tdm_table(timeV + hc, 4u * 41824u);
    agr_tdm_table(disV  + hc, 4u * 50048u);
  }
#endif

  for (int e = tid; e < 512; e += 256) {
    int q = e >> 5, c = e & 31, gq = q0 + q;
    sQ[e] = (gq < 200) ? Qg[(b * 200 + gq) * 64 + hc + c] : 0.f;
  }
  for (int e = tid; e < 208 * 32; e += 256) {
    int r = e >> 5, c = e & 31;
    float kv = 0.f, vv = 0.f;
    if (r < 200) {
      kv = Kg[(b * 200 + r) * 64 + hc + c] + posK[r * 64 + hc + c];
      vv = Vg[(b * 200 + r) * 64 + hc + c] + posV[r * 64 + hc + c];
    }
    sK[e] = kv; sV[e] = vv;
  }
#if AGR_TDM
  if (tid < 32) __builtin_amdgcn_s_wait_tensorcnt(0);
#else
  for (int e = tid; e < 257 * 32; e += 256) {
    int r = e >> 5, c = e & 31;
    sTK[e] = timeK[r * 64 + hc + c];
    sDK[e] = disK [r * 64 + hc + c];
    sTV[e] = timeV[r * 64 + hc + c];
    sDV[e] = disV [r * 64 + hc + c];
  }
#endif
  __syncthreads();

  // qt tables (2 * 16 * 257 dot products of length 32)
  for (int e = tid; e < 2 * 16 * 257; e += 256) {
    int tab = e / (16 * 257);
    int rem = e - tab * (16 * 257);
    int q = rem / 257, t = rem - q * 257;
    const float* T = tab ? sDK : sTK;
    float s = 0.f;
    for (int c = 0; c < 32; ++c) s += sQ[q * 32 + c] * T[t * 32 + c];
    (tab ? sQD : sQT)[q * 257 + t] = s;
  }

  const unsigned wv = tid >> 5, lane = tid & 31;
  const unsigned half = lane >> 4, lo = lane & 15, ko = half * 2;

  // S = Q @ K'^T  (16 x 208) via WMMA, 13 key tiles across 8 waves
  for (int kt = wv; kt < 13; kt += 8) {
    v8f acc = {0.f, 0.f, 0.f, 0.f, 0.f, 0.f, 0.f, 0.f};
    for (int kk = 0; kk < 32; kk += 4) {
      v2f a, bb;
      a.x  = sQ[lo * 32 + kk + ko];
      a.y  = sQ[lo * 32 + kk + ko + 1];
      bb.x = sK[(kt * 16 + lo) * 32 + kk + ko];
      bb.y = sK[(kt * 16 + lo) * 32 + kk + ko + 1];
      acc = wmma4(a, bb, acc);
    }
    for (int v = 0; v < 8; ++v)
      sS[(v + half * 8) * 208 + kt * 16 + lo] = acc[v];
  }
  __syncthreads();

  // add rel score terms, scale, causal + query-pad mask
  for (int e = tid; e < 16 * 208; e += 256) {
    int q = e / 208, k = e - q * 208;
    int gq = q0 + q;
    float s = PADV;
    if (gq < 200 && k <= gq && logseq[b * 200 + gq] != 0) {
      int ti = tmat[(b * 200 + gq) * 200 + k];
      int di = dmat[(b * 200 + gq) * 200 + k];
      s = (sS[e] + sQT[q * 257 + ti] + sQD[q * 257 + di]) * 0.1767766952966369f;
    }
    sS[e] = s;
  }
  __syncthreads();

  // row softmax (16 rows, serial per thread)
  if (tid < 16) {
    float* row = sS + tid * 208;
    float m = row[0];
    for (int k = 1; k < 208; ++k) m = fmaxf(m, row[k]);
    float sum = 0.f;
    for (int k = 0; k < 208; ++k) { float e = __expf(row[k] - m); row[k] = e; sum += e; }
    float inv = 1.f / sum;
    for (int k = 0; k < 208; ++k) row[k] *= inv;
  }
  __syncthreads();

  // O = A @ V'  (16 x 32): waves 0/1 own n-tiles, full K (deterministic)
  if (wv < 2) {
    int nt = wv;
    v8f acc = {0.f, 0.f, 0.f, 0.f, 0.f, 0.f, 0.f, 0.f};
    for (int kk = 0; kk < 208; kk += 4) {
      v2f a, bb;
      a.x  = sS[lo * 208 + kk + ko];
      a.y  = sS[lo * 208 + kk + ko + 1];
      bb.x = sV[(kk + ko) * 32 + nt * 16 + lo];
      bb.y = sV[(kk + ko + 1) * 32 + nt * 16 + lo];
      acc = wmma4(a, bb, acc);
    }
    for (int v = 0; v < 8; ++v)
      sO[(v + half * 8) * 32 + nt * 16 + lo] = acc[v];
  }
  __syncthreads();

  // O += sum_k A[q,k] * (timeV[t[q,k]] + disV[d[q,k]])  — (q, col-pair) per thread
  {
    int q = tid >> 4, cl = tid & 15;
    int gq = q0 + q;
    if (gq < 200) {
      float a0 = 0.f, a1 = 0.f;
      const int* trow = tmat + (b * 200 + gq) * 200;
      const int* drow = dmat + (b * 200 + gq) * 200;
      int kmax = (gq < 199) ? gq : 199;
      for (int k = 0; k <= kmax; ++k) {
        float a = sS[q * 208 + k];
        int ti = trow[k], di = drow[k];
        a0 += a * (sTV[ti * 32 + cl]      + sDV[di * 32 + cl]);
        a1 += a * (sTV[ti * 32 + 16 + cl] + sDV[di * 32 + 16 + cl]);
      }
      sO[q * 32 + cl]      += a0;
      sO[q * 32 + 16 + cl] += a1;
    }
  }
  __syncthreads();

  for (int e = tid; e < 512; e += 256) {
    int q = e >> 5, c = e & 31, gq = q0 + q;
    if (gq < 200) Out[(b * 200 + gq) * 64 + hc + c] = sO[e];
  }
}

// ---------------------------------------------------------------------------
// x = LN(Qn + attn_out); seqs = (x + relu(x@W1+b1)@W2+b2) * rowmask
__global__ __launch_bounds__(256) void agr_ffn(
    const float* __restrict__ Qn, const float* __restrict__ Aout,
    const float* __restrict__ W1, const float* __restrict__ b1,
    const float* __restrict__ W2, const float* __restrict__ b2,
    const float* __restrict__ lns, const float* __restrict__ lnb,
    const int* __restrict__ logseq, float* __restrict__ seqs) {
  extern __shared__ float sm[];
  float* sX  = sm;            // 64x64 (becomes normalized x)
  float* sH  = sm + 4096;     // 64x64 hidden
  float* sW1 = sm + 8192;
  float* sW2 = sm + 12288;
  const int tid = threadIdx.x;
  const int r0 = blockIdx.x * 64;

  for (int e = tid; e < 4096; e += 256) {
    sX[e] = Qn[r0 * 64 + e] + Aout[r0 * 64 + e];
    sW1[e] = W1[e];
    sW2[e] = W2[e];
  }
  __syncthreads();

  if (tid < 64) {
    float mu = 0.f, ss = 0.f;
    for (int c = 0; c < 64; ++c) { float v = sX[tid * 64 + c]; mu += v; ss += v * v; }
    mu *= (1.f / 64.f);
    float var = ss * (1.f / 64.f) - mu * mu;
    float rstd = rsqrtf(var + 1e-8f);
    for (int c = 0; c < 64; ++c)
      sX[tid * 64 + c] = (sX[tid * 64 + c] - mu) * rstd * lns[c] + lnb[c];
  }
  __syncthreads();

  const unsigned wv = tid >> 5, lane = tid & 31;
  const unsigned half = lane >> 4, lo = lane & 15, ko = half * 2;

  for (int t = wv; t < 16; t += 8) {          // H1 = relu(x@W1+b1)
    int tr = t >> 2, tc = t & 3;
    v8f acc = {0.f, 0.f, 0.f, 0.f, 0.f, 0.f, 0.f, 0.f};
    for (int kk = 0; kk < 64; kk += 4) {
      v2f a, b;
      a.x = sX[(tr * 16 + lo) * 64 + kk + ko];
      a.y = sX[(tr * 16 + lo) * 64 + kk + ko + 1];
      b.x = sW1[(kk + ko) * 64 + tc * 16 + lo];
      b.y = sW1[(kk + ko + 1) * 64 + tc * 16 + lo];
      acc = wmma4(a, b, acc);
    }
    for (int v = 0; v < 8; ++v) {
      int lr = tr * 16 + v + half * 8, col = tc * 16 + lo;
      float y = acc[v] + b1[col];
      sH[lr * 64 + col] = (y > 0.f) ? y : 0.f;
    }
  }
  __syncthreads();

  for (int t = wv; t < 16; t += 8) {          // y = H1@W2+b2; out=(x+y)*mask
    int tr = t >> 2, tc = t & 3;
    v8f acc = {0.f, 0.f, 0.f, 0.f, 0.f, 0.f, 0.f, 0.f};
    for (int kk = 0; kk < 64; kk += 4) {
      v2f a, b;
      a.x = sH[(tr * 16 + lo) * 64 + kk + ko];
      a.y = sH[(tr * 16 + lo) * 64 + kk + ko + 1];
      b.x = sW2[(kk + ko) * 64 + tc * 16 + lo];
      b.y = sW2[(kk + ko + 1) * 64 + tc * 16 + lo];
      acc = wmma4(a, b, acc);
    }
    for (int v = 0; v < 8; ++v) {
      int lr = tr * 16 + v + half * 8, col = tc * 16 + lo;
      int g = r0 + lr;
      float y = acc[v] + b2[col] + sX[lr * 64 + col];
      seqs[g * 64 + col] = (logseq[g] != 0) ? y : 0.f;
    }
  }
}

// ---------------------------------------------------------------------------
// log_feats = LN(seqs); pos/neg logits = row dot with gathered item embeddings
__global__ __launch_bounds__(64) void agr_lastln(
    const float* __restrict__ seqs, const float* __restrict__ s,
    const float* __restrict__ bb, const float* __restrict__ item_emb,
    const int* __restrict__ pos_seqs, const int* __restrict__ neg_seqs,
    float* __restrict__ logfeats, float* __restrict__ pos_out,
    float* __restrict__ neg_out) {
  int g = blockIdx.x * 64 + threadIdx.x;   // < 6400
  float x[64];
  float mu = 0.f, ss = 0.f;
  for (int c = 0; c < 64; ++c) { x[c] = seqs[g * 64 + c]; mu += x[c]; ss += x[c] * x[c]; }
  mu *= (1.f / 64.f);
  float var = ss * (1.f / 64.f) - mu * mu;
  float rstd = rsqrtf(var + 1e-8f);
  const float* pe = item_emb + pos_seqs[g] * 64;
  const float* ne = item_emb + neg_seqs[g] * 64;
  float dp = 0.f, dn = 0.f;
  for (int c = 0; c < 64; ++c) {
    float y = (x[c] - mu) * rstd * s[c] + bb[c];
    logfeats[g * 64 + c] = y;
    dp += y * pe[c];
    dn += y * ne[c];
  }
  pos_out[g] = dp;
  neg_out[g] = dn;
}

// ---------------------------------------------------------------------------
// fin_logits = log_feats (6400x64) @ item_emb^T (64x20001), WMMA f32 16x16x4
__global__ __launch_bounds__(256) void agr_fin(const float* __restrict__ lf,
                                               const float* __restrict__ item_emb,
                                               float* __restrict__ out) {
  __shared__ float sA[16 * 64];
  const int tid = threadIdx.x;
  const int r0 = blockIdx.x * 16;
  const int c0 = blockIdx.y * 256;
  for (int e = tid; e < 1024; e += 256) sA[e] = lf[r0 * 64 + e];
  __syncthreads();

  const unsigned wv = tid >> 5, lane = tid & 31;
  const unsigned half = lane >> 4, lo = lane & 15, ko = half * 2;

  for (int t = 0; t < 2; ++t) {
    int item0 = c0 + (wv * 2 + t) * 16;
    int n = item0 + lo;
    bool ok = n < 20001;
    const float* bp = item_emb + (ok ? n : 0) * 64;
    v8f acc = {0.f, 0.f, 0.f, 0.f, 0.f, 0.f, 0.f, 0.f};
    for (int kk = 0; kk < 64; kk += 4) {
      v2f a, b;
      a.x = sA[lo * 64 + kk + ko];
      a.y = sA[lo * 64 + kk + ko + 1];
      b.x = ok ? bp[kk + ko] : 0.f;
      b.y = ok ? bp[kk + ko + 1] : 0.f;
      acc = wmma4(a, b, acc);
    }
    if (ok) {
      for (int v = 0; v < 8; ++v) {
        int row = r0 + v + half * 8;
        out[(size_t)row * 20001 + n] = acc[v];
      }
    }
  }
}

// ---------------------------------------------------------------------------
extern "C" void kernel_launch(void* const* d_in, const int* in_sizes, int n_in,
                              void* d_out, int out_size, void* d_ws, size_t ws_size,
                              hipStream_t stream) {
  const int*   log_seqs = (const int*)  d_in[0];
  const int*   tmat     = (const int*)  d_in[1];
  const int*   dmat     = (const int*)  d_in[2];
  const int*   pos_seqs = (const int*)  d_in[3];
  const int*   neg_seqs = (const int*)  d_in[4];
  const float* item_emb = (const float*)d_in[5];
  const float* posK     = (const float*)d_in[6];
  const float* posV     = (const float*)d_in[7];
  const float* timeK    = (const float*)d_in[8];
  const float* timeV    = (const float*)d_in[9];
  const float* disK     = (const float*)d_in[10];
  const float* disV     = (const float*)d_in[11];
  const float* Wq       = (const float*)d_in[12];
  const float* bq       = (const float*)d_in[13];
  const float* Wk       = (const float*)d_in[14];
  const float* bk       = (const float*)d_in[15];
  const float* Wv       = (const float*)d_in[16];
  const float* bv       = (const float*)d_in[17];
  const float* ln_as    = (const float*)d_in[18];
  const float* ln_ab    = (const float*)d_in[19];
  const float* ln_fs    = (const float*)d_in[20];
  const float* ln_fb    = (const float*)d_in[21];
  const float* W1       = (const float*)d_in[22];
  const float* b1       = (const float*)d_in[23];
  const float* W2       = (const float*)d_in[24];
  const float* b2       = (const float*)d_in[25];
  const float* lls      = (const float*)d_in[26];
  const float* llb      = (const float*)d_in[27];

  const int R = 32 * 200;                // 6400 rows
  float* ws    = (float*)d_ws;
  float* seqs  = ws;
  float* Qn    = seqs + R * 64;
  float* Qb    = Qn   + R * 64;
  float* Kb    = Qb   + R * 64;
  float* Vb    = Kb   + R * 64;
  float* Aout  = Vb   + R * 64;
  float* lfeat = Aout + R * 64;          // total ~11.5 MB

  float* pos_out = (float*)d_out;        // (B,L)
  float* neg_out = pos_out + R;          // (B,L)
  float* fin_out = neg_out + R;          // (B*L, 20001)

  agr_embed<<<1600, 256, 0, stream>>>(log_seqs, item_emb, seqs);

  const size_t qkv_lds  = 20480u * sizeof(float);   //  80 KB
  const size_t attn_lds = 58784u * sizeof(float);   // ~235 KB (fits 320 KB WGP LDS)
  const size_t ffn_lds  = 16384u * sizeof(float);   //  64 KB

  for (int i = 0; i < 2; ++i) {
    agr_qkv<<<100, 256, qkv_lds, stream>>>(
        seqs, Qn, Qb, Kb, Vb,
        Wq + i * 4096, bq + i * 64, Wk + i * 4096, bk + i * 64,
        Wv + i * 4096, bv + i * 64, ln_as + i * 64, ln_ab + i * 64);
    agr_attn<<<32 * 2 * 13, 256, attn_lds, stream>>>(
        Qb, Kb, Vb, posK, posV, timeK, timeV, disK, disV,
        tmat, dmat, log_seqs, Aout);
    agr_ffn<<<100, 256, ffn_lds, stream>>>(
        Qn, Aout, W1 + i * 4096, b1 + i * 64, W2 + i * 4096, b2 + i * 64,
        ln_fs + i * 64, ln_fb + i * 64, log_seqs, seqs);
  }

  agr_lastln<<<100, 64, 0, stream>>>(seqs, lls, llb, item_emb, pos_seqs,
                                     neg_seqs, lfeat, pos_out, neg_out);

  dim3 fgrid(400, 79);                   // 6400/16 rows, ceil(20001/256) item blocks
  agr_fin<<<fgrid, 256, 0, stream>>>(lfeat, item_emb, fin_out);
}